// EdgeLoss_32452772889049
// MI455X (gfx1250) — compile-verified
//
#include <hip/hip_runtime.h>
#include <math.h>

// ---------------------------------------------------------------------------
// EdgeLoss for MI455X (gfx1250, wave32).
//
// vec:   (1,16,20,256,256) f32   label: (1,1,20,256,256) i32   mask: f32 ones
// out:   [loss, nmsk] f32
//
// L2-resident workload (94 MB < 192 MB L2); ~5.5 GB of L2 reads total ->
// L2-bandwidth bound. Design goals:
//   * fully coalesced x-contiguous b32 loads in one big clause per thread
//     (no predication branches: indices are clamped, invalid lanes zeroed
//     with cndmask) -> single s_wait_loadcnt, maximal MLP
//   * 32-bit per-lane indices off a uniform SGPR base -> GVS addressing,
//     no per-load 64-bit VALU address adds
//   * channel-dot (K=16) as WMMA row-sum: B = elementwise products
//     (4 channels x 16 positions), A = ones, 4 accumulating
//     v_wmma_f32_16x16x4_f32 per tile; every row of D equals the column sum,
//     so D[0] holds the logit for the lane's own column (no shuffles), and
//     the XDL pipe co-executes with the VALU product/BCE work.
// ---------------------------------------------------------------------------

typedef __attribute__((ext_vector_type(2))) float v2f;
typedef __attribute__((ext_vector_type(8))) float v8f;

#define ZDIM 20
#define YDIM 256
#define XDIM 256
#define CDIM 16
#define CSTRI 1310720 /* channel stride in elements: 20*256*256 */
#define ZSTRI 65536   /* 256*256 */
#define YSTRI 256
#define N_EDGE 32

struct EdgeTable {
  int oz[N_EDGE];
  int oy[N_EDGE];
  int ox[N_EDGE];
};

// ------------------------------- device ------------------------------------

__global__ void zero_out_kernel(float* __restrict__ out, int n) {
  int i = blockIdx.x * blockDim.x + threadIdx.x;
  if (i < n) out[i] = 0.0f;
}

__global__ __launch_bounds__(256) void edge_affinity_loss_kernel(
    const float* __restrict__ vec, const int* __restrict__ label,
    const float* __restrict__ mask, float* __restrict__ out, EdgeTable et) {
  const int e = blockIdx.z;
  const int z = blockIdx.y;
  const int y = blockIdx.x;

  const int oz = et.oz[e], oy = et.oy[e], ox = et.ox[e];
  const int az = oz < 0 ? -oz : oz;
  const int ay = oy < 0 ? -oy : oy;
  const int ax = ox < 0 ? -ox : ox;
  const int CZ = ZDIM - az, CY = YDIM - ay, CX = XDIM - ax;

  if (z >= CZ || y >= CY) return;  // uniform per block: EXEC stays all-ones

  // crop-1 / crop-2 base offsets (x origin folded into base); bases are
  // uniform -> SGPR base + 32-bit VGPR element index (GVS addressing).
  const int z1 = z + (oz > 0 ? oz : 0), y1 = y + (oy > 0 ? oy : 0);
  const int z2 = z + (oz < 0 ? -oz : 0), y2 = y + (oy < 0 ? -oy : 0);
  const int b1 = z1 * ZSTRI + y1 * YSTRI + (ox > 0 ? ox : 0);
  const int b2 = z2 * ZSTRI + y2 * YSTRI + (ox < 0 ? -ox : 0);
  const float* __restrict__ v1p = vec + b1;
  const float* __restrict__ v2p = vec + b2;

  const int tid = threadIdx.x;
  const int wv = tid >> 5;       // wave id 0..7, owns x tile [32*wv, 32*wv+32)
  const int lane = tid & 31;
  const int pos = lane & 15;     // column index within a 16-position tile
  const int hi = lane >> 4;      // half-wave selects channel pair within K=4

  const int xA = wv * 32 + pos;  // tile A positions
  const int xB = xA + 16;        // tile B positions
  const bool vA = xA < CX;
  const bool vB = xB < CX;
  // Clamp (CX >= 225 > 0 always) so loads run with full EXEC -> one clause.
  const int xAc = vA ? xA : CX - 1;
  const int xBc = vB ? xB : CX - 1;
  const int offA = 2 * hi * CSTRI + xAc;  // per-lane 32-bit element index
  const int offB = 2 * hi * CSTRI + xBc;

  // ---- batched loads: 32 b32 values, no control flow in between ----------
  float a1[8], a2[8], b1v[8], b2v[8];
#pragma unroll
  for (int i = 0; i < 4; ++i) {
    const int c0 = 4 * i * CSTRI;  // uniform per unrolled step
    const int c1 = c0 + CSTRI;
    a1[2 * i + 0] = v1p[c0 + offA];
    a2[2 * i + 0] = v2p[c0 + offA];
    a1[2 * i + 1] = v1p[c1 + offA];
    a2[2 * i + 1] = v2p[c1 + offA];
    b1v[2 * i + 0] = v1p[c0 + offB];
    b2v[2 * i + 0] = v2p[c0 + offB];
    b1v[2 * i + 1] = v1p[c1 + offB];
    b2v[2 * i + 1] = v2p[c1 + offB];
  }

  // ---- products (invalid lanes zeroed via cndmask, no branches) ----------
  const float zA = vA ? 1.0f : 0.0f;
  const float zB = vB ? 1.0f : 0.0f;
  const v2f ones = {1.0f, 1.0f};  // A matrix: 16x4 all-ones -> D = column sums
  v8f dA = {};
  v8f dB = {};
#pragma unroll
  for (int i = 0; i < 4; ++i) {
    v2f pA, pB;
    pA.x = zA * (a1[2 * i + 0] * a2[2 * i + 0]);
    pA.y = zA * (a1[2 * i + 1] * a2[2 * i + 1]);
    pB.x = zB * (b1v[2 * i + 0] * b2v[2 * i + 0]);
    pB.y = zB * (b1v[2 * i + 1] * b2v[2 * i + 1]);
    // D += ones(16x4) x B(4x16): accumulates 4 channels of the dot per step.
    dA = __builtin_amdgcn_wmma_f32_16x16x4_f32(false, ones, false, pA,
                                               (short)0, dA, false, false);
    dB = __builtin_amdgcn_wmma_f32_16x16x4_f32(false, ones, false, pB,
                                               (short)0, dB, false, false);
  }

  // D row-0 layout: lanes 0-15 -> (M=0,N=lane); lanes 16-31 -> (M=8,N=lane-16).
  // All rows equal => dA[0] = pred at column (lane&15). Lane `l` owns position
  // wv*32 + l: tile A for lanes 0-15, tile B for lanes 16-31.
  const float pred = (lane < 16) ? dA[0] : dB[0];
  const int xp = wv * 32 + lane;
  const bool vp = xp < CX;
  const int xpc = vp ? xp : CX - 1;

  // ---- target / weight / BCE (clamped loads, selects instead of branch) --
  const int l1 = label[b1 + xpc];
  const int l2 = label[b2 + xpc];
  const float m1 = mask[b1 + xpc];
  const float m2 = mask[b2 + xpc];
  float w = m1 * (l1 != 0 ? 1.0f : 0.0f) * m2 * (l2 != 0 ? 1.0f : 0.0f);
  w = vp ? w : 0.0f;
  const float tgt = (l1 == l2) ? 1.0f : 0.0f;
  // stable BCE-with-logits: max(p,0) - p*t + log1p(exp(-|p|))
  const float bce =
      fmaxf(pred, 0.0f) - pred * tgt + log1pf(__expf(-fabsf(pred)));
  const float lossAcc = w * bce;
  const float nmAcc = (w > 0.0f) ? 1.0f : 0.0f;

  // ---- block tree reduction in LDS, one float atomic per output ----------
  __shared__ float sred[2][256];
  sred[0][tid] = lossAcc;
  sred[1][tid] = nmAcc;
  __syncthreads();
#pragma unroll
  for (int s = 128; s > 0; s >>= 1) {
    if (tid < s) {
      sred[0][tid] += sred[0][tid + s];
      sred[1][tid] += sred[1][tid + s];
    }
    __syncthreads();
  }
  if (tid == 0) {
    __hip_atomic_fetch_add(&out[0], sred[0][0], __ATOMIC_RELAXED,
                           __HIP_MEMORY_SCOPE_AGENT);
    __hip_atomic_fetch_add(&out[1], sred[1][0], __ATOMIC_RELAXED,
                           __HIP_MEMORY_SCOPE_AGENT);
  }
}

// -------------------------------- host -------------------------------------
// Exact replication of np.random.RandomState(0) edge sampling:
// MT19937 with Knuth init; legacy randint = masked rejection on 64-bit draws
// where next64 = (next32 << 32) | next32. Ranges here are powers of two, so
// exactly 6 draws per edge, no rejection.

namespace {

struct MT19937 {
  unsigned mt[624];
  int mti;
  explicit MT19937(unsigned s) {
    mt[0] = s;
    for (int i = 1; i < 624; ++i)
      mt[i] = 1812433253u * (mt[i - 1] ^ (mt[i - 1] >> 30)) + (unsigned)i;
    mti = 624;
  }
  unsigned next32() {
    if (mti >= 624) {
      for (int i = 0; i < 624; ++i) {
        unsigned y = (mt[i] & 0x80000000u) | (mt[(i + 1) % 624] & 0x7fffffffu);
        mt[i] = mt[(i + 397) % 624] ^ (y >> 1) ^ ((y & 1u) ? 0x9908b0dfu : 0u);
      }
      mti = 0;
    }
    unsigned y = mt[mti++];
    y ^= y >> 11;
    y ^= (y << 7) & 0x9d2c5680u;
    y ^= (y << 15) & 0xefc60000u;
    y ^= y >> 18;
    return y;
  }
  unsigned long long next64() {
    unsigned long long h = next32();
    unsigned long long l = next32();
    return (h << 32) | l;
  }
  int randint(unsigned long long n) {  // numpy legacy randint(0, n)
    unsigned long long rng = n - 1ull;
    unsigned long long msk = rng;
    msk |= msk >> 1;
    msk |= msk >> 2;
    msk |= msk >> 4;
    msk |= msk >> 8;
    msk |= msk >> 16;
    msk |= msk >> 32;
    for (;;) {
      unsigned long long v = next64() & msk;
      if (v <= rng) return (int)v;
    }
  }
};

EdgeTable make_edges() {
  EdgeTable et;
  MT19937 rs(0u);
  for (int e = 0; e < N_EDGE; ++e) {
    const int x = rs.randint(32);
    const int y = rs.randint(32);
    const int z = rs.randint(16);
    const int sz = rs.randint(2) ? -1 : 1;  // rs.choice([1,-1]) order: z, y, x
    const int sy = rs.randint(2) ? -1 : 1;
    const int sx = rs.randint(2) ? -1 : 1;
    et.oz[e] = z * sz;
    et.oy[e] = y * sy;
    et.ox[e] = x * sx;
  }
  return et;
}

}  // namespace

extern "C" void kernel_launch(void* const* d_in, const int* in_sizes, int n_in,
                              void* d_out, int out_size, void* d_ws,
                              size_t ws_size, hipStream_t stream) {
  (void)in_sizes;
  (void)n_in;
  (void)d_ws;
  (void)ws_size;
  const float* vec = (const float*)d_in[0];
  const int* label = (const int*)d_in[1];
  const float* mask = (const float*)d_in[2];
  float* out = (float*)d_out;

  const EdgeTable et = make_edges();

  zero_out_kernel<<<1, 64, 0, stream>>>(out, out_size);

  dim3 grid(YDIM, ZDIM, N_EDGE);
  edge_affinity_loss_kernel<<<grid, 256, 0, stream>>>(vec, label, mask, out, et);
}